// TCR_28948079575516
// MI455X (gfx1250) — compile-verified
//
#include <hip/hip_runtime.h>
#include <math.h>

// Problem dims (fixed by the reference)
#define NROWS 4096      // topic_emb rows
#define NCOLS 2048      // cluster_center rows
#define DIM   256
#define ALPHA 0.05f
#define INV_ALPHA 20.0f
#define EPS 1e-16f
#define STOP_THR 0.005f
#define MAX_ITER 500

typedef float v2f __attribute__((ext_vector_type(2)));
typedef float v8f __attribute__((ext_vector_type(8)));

// ---------------------------------------------------------------------------
// init: u = 1/n, v = 1/m, flag = 0, errAcc = 0, out = 0
// ---------------------------------------------------------------------------
__global__ void tcr_init(float* u, float* v, float* errAcc, int* flag, float* out) {
    int i = blockIdx.x * blockDim.x + threadIdx.x;
    if (i < NROWS) u[i] = 1.0f / (float)NROWS;
    if (i < NCOLS) v[i] = 1.0f / (float)NCOLS;
    if (i == 0) { *flag = 0; *errAcc = 0.0f; *out = 0.0f; }
}

// ---------------------------------------------------------------------------
// squared row norms of X (4096 rows) and Y (2048 rows); one wave per row
// ---------------------------------------------------------------------------
__global__ void tcr_sqnorms(const float* __restrict__ X, const float* __restrict__ Y,
                            float* __restrict__ x2, float* __restrict__ y2) {
    int wid  = (blockIdx.x * blockDim.x + threadIdx.x) >> 5;   // global wave id
    int lane = threadIdx.x & 31;
    const float* row;
    if (wid < NROWS)               row = X + (size_t)wid * DIM;
    else if (wid < NROWS + NCOLS)  row = Y + (size_t)(wid - NROWS) * DIM;
    else return;
    float s = 0.0f;
    #pragma unroll
    for (int t = 0; t < DIM / 32; ++t) {
        float x = row[lane + 32 * t];
        s += x * x;
    }
    #pragma unroll
    for (int off = 16; off > 0; off >>= 1) s += __shfl_xor(s, off, 32);
    if (lane == 0) {
        if (wid < NROWS) x2[wid] = s;
        else             y2[wid - NROWS] = s;
    }
}

// ---------------------------------------------------------------------------
// K = exp(-alpha * sqrt(max(x2 + y2 - 2*X.Y^T, 0)))  via WMMA f32 16x16x4.
// Each wave computes a 16(M) x 64(N) strip: 1 A-fragment reused over 4 B tiles.
// Strips: (4096/16) * (2048/64) = 256 * 32 = 8192; 8 waves/block -> 1024 blocks.
// ---------------------------------------------------------------------------
__global__ void tcr_kernelmat(const float* __restrict__ X, const float* __restrict__ Y,
                              const float* __restrict__ x2, const float* __restrict__ y2,
                              float* __restrict__ K) {
    int wid     = blockIdx.x * (blockDim.x >> 5) + (threadIdx.x >> 5);
    int strip_m = wid >> 5;          // 0..255
    int strip_n = wid & 31;          // 0..31
    int rowBase = strip_m * 16;
    int colBase = strip_n * 64;

    int lane = threadIdx.x & 31;
    int lidx = lane & 15;            // M (for A) / N (for B) within tile
    int kOff = (lane >> 4) * 2;      // lanes 0-15 -> K=0,1 ; lanes 16-31 -> K=2,3

    v8f acc0 = {}, acc1 = {}, acc2 = {}, acc3 = {};

    const float* arow = X + (size_t)(rowBase + lidx) * DIM + kOff;
    const float* b0   = Y + (size_t)(colBase +  0 + lidx) * DIM + kOff;
    const float* b1   = Y + (size_t)(colBase + 16 + lidx) * DIM + kOff;
    const float* b2   = Y + (size_t)(colBase + 32 + lidx) * DIM + kOff;
    const float* b3   = Y + (size_t)(colBase + 48 + lidx) * DIM + kOff;

    #pragma unroll 4
    for (int k = 0; k < DIM; k += 4) {
        v2f a  = *(const v2f*)(arow + k);
        v2f f0 = *(const v2f*)(b0 + k);
        v2f f1 = *(const v2f*)(b1 + k);
        v2f f2 = *(const v2f*)(b2 + k);
        v2f f3 = *(const v2f*)(b3 + k);
        acc0 = __builtin_amdgcn_wmma_f32_16x16x4_f32(false, a, false, f0, (short)0, acc0, false, false);
        acc1 = __builtin_amdgcn_wmma_f32_16x16x4_f32(false, a, false, f1, (short)0, acc1, false, false);
        acc2 = __builtin_amdgcn_wmma_f32_16x16x4_f32(false, a, false, f2, (short)0, acc2, false, false);
        acc3 = __builtin_amdgcn_wmma_f32_16x16x4_f32(false, a, false, f3, (short)0, acc3, false, false);
    }

    // C/D layout: VGPR r, lanes 0-15 -> M=r ; lanes 16-31 -> M=8+r ; N = lane&15
    int nIdx = lane & 15;
    int mHalf = (lane >> 4) * 8;
    #pragma unroll
    for (int r = 0; r < 8; ++r) {
        int row = rowBase + mHalf + r;
        float xr = x2[row];
        float dots[4] = { acc0[r], acc1[r], acc2[r], acc3[r] };
        #pragma unroll
        for (int t = 0; t < 4; ++t) {
            int col = colBase + t * 16 + nIdx;
            float d2 = xr + y2[col] - 2.0f * dots[t];
            float M  = sqrtf(fmaxf(d2, 0.0f));
            K[(size_t)row * NCOLS + col] = expf(-ALPHA * M);
        }
    }
}

// ---------------------------------------------------------------------------
// v = b / (K^T u + eps)   — one thread per column (coalesced down K's rows)
// ---------------------------------------------------------------------------
__global__ void tcr_sink_v(const float* __restrict__ K, const float* __restrict__ u,
                           float* __restrict__ v, const int* __restrict__ flag) {
    if (*flag) return;
    int j = blockIdx.x * blockDim.x + threadIdx.x;   // < NCOLS
    const float* col = K + j;
    float s = 0.0f;
    #pragma unroll 4
    for (int i = 0; i < NROWS; ++i)
        s += col[(size_t)i * NCOLS] * u[i];
    v[j] = (1.0f / (float)NCOLS) / (s + EPS);
}

// ---------------------------------------------------------------------------
// u = a / (K v + eps)   — one wave per row + shuffle reduction
// ---------------------------------------------------------------------------
__global__ void tcr_sink_u(const float* __restrict__ K, const float* __restrict__ v,
                           float* __restrict__ u, const int* __restrict__ flag) {
    if (*flag) return;
    int row  = (blockIdx.x * blockDim.x + threadIdx.x) >> 5;  // < NROWS
    int lane = threadIdx.x & 31;
    const float* krow = K + (size_t)row * NCOLS;
    float s = 0.0f;
    #pragma unroll 4
    for (int j = lane; j < NCOLS; j += 32)
        s += krow[j] * v[j];
    #pragma unroll
    for (int off = 16; off > 0; off >>= 1) s += __shfl_xor(s, off, 32);
    if (lane == 0) u[row] = (1.0f / (float)NROWS) / (s + EPS);
}

// ---------------------------------------------------------------------------
// convergence check:  err = sum_j | v_j * (K^T u)_j - 1/m |
// ---------------------------------------------------------------------------
__global__ void tcr_zero_err(float* errAcc, const int* flag) {
    if (*flag) return;
    *errAcc = 0.0f;
}

__global__ void tcr_err(const float* __restrict__ K, const float* __restrict__ u,
                        const float* __restrict__ v, float* errAcc,
                        const int* __restrict__ flag) {
    if (*flag) return;
    int j = blockIdx.x * blockDim.x + threadIdx.x;   // < NCOLS
    const float* col = K + j;
    float s = 0.0f;
    #pragma unroll 4
    for (int i = 0; i < NROWS; ++i)
        s += col[(size_t)i * NCOLS] * u[i];
    float e = fabsf(v[j] * s - 1.0f / (float)NCOLS);
    __shared__ float red[256];
    red[threadIdx.x] = e;
    __syncthreads();
    for (int off = 128; off > 0; off >>= 1) {
        if ((int)threadIdx.x < off) red[threadIdx.x] += red[threadIdx.x + off];
        __syncthreads();
    }
    if (threadIdx.x == 0) atomicAdd(errAcc, red[0]);
}

__global__ void tcr_set_flag(const float* errAcc, int* flag) {
    if (*flag) return;
    if (*errAcc <= STOP_THR) *flag = 1;
}

// ---------------------------------------------------------------------------
// loss = sum_ij u_i * K_ij * v_j * M_ij,  with M_ij = -ln(K_ij)/alpha
// ---------------------------------------------------------------------------
__global__ void tcr_loss(const float* __restrict__ K, const float* __restrict__ u,
                         const float* __restrict__ v, float* out) {
    const size_t total = (size_t)NROWS * NCOLS;
    size_t stride = (size_t)gridDim.x * blockDim.x;
    float acc = 0.0f;
    for (size_t p = blockIdx.x * (size_t)blockDim.x + threadIdx.x; p < total; p += stride) {
        float kv = K[p];
        int i = (int)(p >> 11);          // p / 2048
        int j = (int)(p & (NCOLS - 1));  // p % 2048
        float M = -logf(kv) * INV_ALPHA;
        acc += u[i] * kv * v[j] * M;
    }
    __shared__ float red[256];
    red[threadIdx.x] = acc;
    __syncthreads();
    for (int off = 128; off > 0; off >>= 1) {
        if ((int)threadIdx.x < off) red[threadIdx.x] += red[threadIdx.x + off];
        __syncthreads();
    }
    if (threadIdx.x == 0) atomicAdd(out, red[0]);
}

// ---------------------------------------------------------------------------
extern "C" void kernel_launch(void* const* d_in, const int* in_sizes, int n_in,
                              void* d_out, int out_size, void* d_ws, size_t ws_size,
                              hipStream_t stream) {
    const float* X = (const float*)d_in[0];   // topic_emb   [4096,256]
    const float* Y = (const float*)d_in[1];   // cluster_ctr [2048,256]
    float* out = (float*)d_out;

    float* ws   = (float*)d_ws;
    float* K    = ws;                                   // 8M floats = 32 MB
    float* x2   = ws + (size_t)NROWS * NCOLS;           // 4096
    float* y2   = x2 + NROWS;                           // 2048
    float* u    = y2 + NCOLS;                           // 4096
    float* v    = u + NROWS;                            // 2048
    float* errA = v + NCOLS;                            // 1
    int*   flag = (int*)(errA + 1);                     // 1

    tcr_init<<<16, 256, 0, stream>>>(u, v, errA, flag, out);
    tcr_sqnorms<<<(NROWS + NCOLS) / 8, 256, 0, stream>>>(X, Y, x2, y2);
    tcr_kernelmat<<<1024, 256, 0, stream>>>(X, Y, x2, y2, K);

    for (int cpt = 1; cpt <= MAX_ITER; ++cpt) {
        tcr_sink_v<<<NCOLS / 256, 256, 0, stream>>>(K, u, v, flag);
        tcr_sink_u<<<NROWS * 32 / 256, 256, 0, stream>>>(K, v, u, flag);
        if (cpt % 50 == 1) {
            tcr_zero_err<<<1, 1, 0, stream>>>(errA, flag);
            tcr_err<<<NCOLS / 256, 256, 0, stream>>>(K, u, v, errA, flag);
            tcr_set_flag<<<1, 1, 0, stream>>>(errA, flag);
        }
    }

    tcr_loss<<<512, 256, 0, stream>>>(K, u, v, out);
}